// OCLeNet_266287972883
// MI455X (gfx1250) — compile-verified
//
#include <hip/hip_runtime.h>
#include <hip/hip_bf16.h>

typedef __attribute__((ext_vector_type(2))) float v2f;
typedef __attribute__((ext_vector_type(8))) float v8f;

// ---------------------------------------------------------------------------
// Stage GEMM: y[N, COUT] = gather(x, neigh)[N, 27*CIN] @ W[27*CIN, COUT]
// One wave computes a 16-row M tile x all COUT columns using
// V_WMMA_F32_16X16X4_F32 (fp32 matrix core, full reference precision).
// Block = 128 threads = 4 waves = 64 rows.
// Also emits per-block per-channel sum/sumsq partials (deterministic, no atomics).
// ---------------------------------------------------------------------------
template <int CIN, int COUT>
__global__ __launch_bounds__(128)
void gather_gemm_wmma(const float* __restrict__ x,      // [Nin, CIN]
                      const int*   __restrict__ neigh,  // [N, 27]
                      const float* __restrict__ W,      // [27*CIN, COUT]
                      float* __restrict__ y,            // [N, COUT]
                      float* __restrict__ part)         // [gridDim.x, 2, COUT]
{
    static_assert(CIN % 4 == 0 && COUT % 16 == 0, "shape");
    constexpr int NT = COUT / 16;

    const int lane    = threadIdx.x & 31;
    const int wave    = threadIdx.x >> 5;
    const int half    = lane >> 4;       // 0 -> K {0,1}, 1 -> K {2,3} of each step
    const int l       = lane & 15;       // M row (A) / N col (B) within tile
    const int rowbase = blockIdx.x * 64 + wave * 16;
    const int row     = rowbase + l;

    v8f acc[NT] = {};

    const int* nrow = neigh + (long)row * 27;
    for (int j = 0; j < 27; ++j) {
        const int idx = nrow[j];
        const float* xr = x + (long)idx * CIN;
        for (int c0 = 0; c0 < CIN; c0 += 4) {
            // A fragment: lane(l,half) holds g[row][k0+half*2 + {0,1}]
            const v2f a = *reinterpret_cast<const v2f*>(xr + c0 + half * 2);
            const int kg = j * CIN + c0 + half * 2;   // W row for b.x
            const float* wrow = W + (long)kg * COUT;
#pragma unroll
            for (int t = 0; t < NT; ++t) {
                const int col = t * 16 + l;
                v2f b;
                b.x = wrow[col];
                b.y = wrow[COUT + col];
                acc[t] = __builtin_amdgcn_wmma_f32_16x16x4_f32(
                    /*neg_a=*/false, a, /*neg_b=*/false, b,
                    /*c_mod=*/(short)0, acc[t],
                    /*reuse_a=*/false, /*reuse_b=*/false);
            }
        }
    }

    // Writeback + per-channel sum/sumsq (deterministic reduction).
    __shared__ float s_sum[4][COUT];
    __shared__ float s_sq[4][COUT];

#pragma unroll
    for (int t = 0; t < NT; ++t) {
        float s = 0.f, sq = 0.f;
#pragma unroll
        for (int r = 0; r < 8; ++r) {
            // C/D layout: VGPR r, lanes 0-15 -> M=r, lanes 16-31 -> M=8+r
            const float v = acc[t][r];
            y[(long)(rowbase + half * 8 + r) * COUT + t * 16 + l] = v;
            s += v;
            sq += v * v;
        }
        s  += __shfl_xor(s, 16);
        sq += __shfl_xor(sq, 16);
        if (half == 0) {
            s_sum[wave][t * 16 + l] = s;
            s_sq[wave][t * 16 + l]  = sq;
        }
    }
    __syncthreads();
    if (threadIdx.x < COUT) {
        float s = 0.f, sq = 0.f;
#pragma unroll
        for (int w = 0; w < 4; ++w) { s += s_sum[w][threadIdx.x]; sq += s_sq[w][threadIdx.x]; }
        float* p = part + (long)blockIdx.x * 2 * COUT;
        p[threadIdx.x]        = s;
        p[COUT + threadIdx.x] = sq;
    }
}

// Reduce block partials in fixed order -> BN scale/shift per channel.
__global__ void bn_finalize(const float* __restrict__ part, int nblocks, int cout,
                            const float* __restrict__ gamma, const float* __restrict__ beta,
                            float invN, float* __restrict__ stats)
{
    const int c = threadIdx.x;
    if (c >= cout) return;
    float s = 0.f, sq = 0.f;
    for (int b = 0; b < nblocks; ++b) {
        s  += part[(long)b * 2 * cout + c];
        sq += part[(long)b * 2 * cout + cout + c];
    }
    const float mean = s * invN;
    const float var  = sq * invN - mean * mean;
    const float sc   = gamma[c] * rsqrtf(var + 1e-5f);
    stats[c]        = sc;
    stats[cout + c] = beta[c] - mean * sc;
}

// Affine (BN) per element -> max over 8 siblings -> ReLU (commutes with max).
__global__ void bn_relu_pool(const float* __restrict__ y, const float* __restrict__ stats,
                             float* __restrict__ out, int nparent, int cout)
{
    const int tid = blockIdx.x * blockDim.x + threadIdx.x;
    if (tid >= nparent * cout) return;
    const int c = tid % cout;
    const int p = tid / cout;
    const float sc = stats[c];
    const float sh = stats[cout + c];
    float m = -3.402823466e+38f;
    const float* src = y + (long)p * 8 * cout + c;
#pragma unroll
    for (int s = 0; s < 8; ++s) {
        const float v = src[(long)s * cout] * sc + sh;
        m = fmaxf(m, v);
    }
    out[tid] = fmaxf(m, 0.f);
}

__global__ void zero_f32(float* __restrict__ p, int n)
{
    const int tid = blockIdx.x * blockDim.x + threadIdx.x;
    if (tid < n) p[tid] = 0.f;
}

// octree2voxel: vox[b, c, h, h, h] with v = vox_idx[node], b = v/512, rem = v%512
__global__ void scatter_vox(const float* __restrict__ x, const int* __restrict__ vidx,
                            float* __restrict__ out, int n, int cout)
{
    const int tid = blockIdx.x * blockDim.x + threadIdx.x;
    if (tid >= n * cout) return;
    const int c = tid % cout;
    const int node = tid / cout;
    const int v = vidx[node];
    const int b = v >> 9;         // / 512
    const int rem = v & 511;      // % 512
    out[((long)b * cout + c) * 512 + rem] = x[(long)node * cout + c];
}

extern "C" void kernel_launch(void* const* d_in, const int* in_sizes, int n_in,
                              void* d_out, int out_size, void* d_ws, size_t ws_size,
                              hipStream_t stream)
{
    const float* data   = (const float*)d_in[0];
    const float* W0     = (const float*)d_in[1];
    const float* g0     = (const float*)d_in[2];
    const float* b0     = (const float*)d_in[3];
    const float* W1     = (const float*)d_in[4];
    const float* g1     = (const float*)d_in[5];
    const float* b1     = (const float*)d_in[6];
    const float* W2     = (const float*)d_in[7];
    const float* g2     = (const float*)d_in[8];
    const float* b2     = (const float*)d_in[9];
    const int*   neigh0 = (const int*)d_in[10];
    const int*   neigh1 = (const int*)d_in[11];
    const int*   neigh2 = (const int*)d_in[12];
    const int*   vidx   = (const int*)d_in[13];

    const int N6 = in_sizes[10] / 27;   // 524288
    const int N5 = in_sizes[11] / 27;   // 65536
    const int N4 = in_sizes[12] / 27;   // 8192
    const int N3 = in_sizes[13];        // 1024

    char* ws = (char*)d_ws;
    float* Y     = (float*)(ws);                                   // 32 MB (reused each stage)
    float* P0    = (float*)(ws + 33554432);                        // 4 MB
    float* P1    = (float*)(ws + 33554432 + 4194304);              // 1 MB
    float* P2    = (float*)(ws + 33554432 + 4194304 + 1048576);    // 256 KB
    float* part  = (float*)(ws + 33554432 + 4194304 + 1048576 + 262144);           // 1 MB
    float* stats = (float*)(ws + 33554432 + 4194304 + 1048576 + 262144 + 1048576); // 512 B

    float* out = (float*)d_out;
    zero_f32<<<(out_size + 255) / 256, 256, 0, stream>>>(out, out_size);

    // Stage 0: [N6,4] -> [N6,16] -> pool [N5,16]
    const int nb0 = N6 / 64;
    gather_gemm_wmma<4, 16><<<nb0, 128, 0, stream>>>(data, neigh0, W0, Y, part);
    bn_finalize<<<1, 64, 0, stream>>>(part, nb0, 16, g0, b0, 1.f / (float)N6, stats);
    bn_relu_pool<<<(N5 * 16 + 255) / 256, 256, 0, stream>>>(Y, stats, P0, N5, 16);

    // Stage 1: [N5,16] -> [N5,32] -> pool [N4,32]
    const int nb1 = N5 / 64;
    gather_gemm_wmma<16, 32><<<nb1, 128, 0, stream>>>(P0, neigh1, W1, Y, part);
    bn_finalize<<<1, 64, 0, stream>>>(part, nb1, 32, g1, b1, 1.f / (float)N5, stats);
    bn_relu_pool<<<(N4 * 32 + 255) / 256, 256, 0, stream>>>(Y, stats, P1, N4, 32);

    // Stage 2: [N4,32] -> [N4,64] -> pool [N3,64]
    const int nb2 = N4 / 64;
    gather_gemm_wmma<32, 64><<<nb2, 128, 0, stream>>>(P1, neigh2, W2, Y, part);
    bn_finalize<<<1, 64, 0, stream>>>(part, nb2, 64, g2, b2, 1.f / (float)N4, stats);
    bn_relu_pool<<<(N3 * 64 + 255) / 256, 256, 0, stream>>>(Y, stats, P2, N3, 64);

    // octree2voxel scatter + transpose
    scatter_vox<<<(N3 * 64 + 255) / 256, 256, 0, stream>>>(P2, vidx, out, N3, 64);
}